// Attention_7988639170755
// MI455X (gfx1250) — compile-verified
//
#include <hip/hip_runtime.h>
#include <math.h>

// Problem constants (match reference).
#define BB 8
#define OO 128
#define II 128
#define DD 1024

typedef float v2f __attribute__((ext_vector_type(2)));
typedef float v8f __attribute__((ext_vector_type(8)));

// V_WMMA_F32_16X16X4_F32: A=16x4 f32 (2 VGPRs), B=4x16 f32 (2 VGPRs), C/D=16x16 f32 (8 VGPRs).
// 8-arg form: (neg_a, A, neg_b, B, c_mod, C, reuse_a, reuse_b)
__device__ __forceinline__ v8f wmma4(v2f a, v2f b, v8f c) {
  return __builtin_amdgcn_wmma_f32_16x16x4_f32(false, a, false, b, (short)0, c,
                                               false, false);
}

// ---------------------------------------------------------------------------
// Generic f32 WMMA GEMM:  C[M,N] = A[M,K] @ W[K,N]   (row-major, batched via z)
// blockDim.x = 128 (4 waves); each wave computes one 16x16 tile of C.
// A-layout per lane: (M = lane&15, K = k + 2*(lane>>4) + {0,1})  -> one b64 load
// B-layout per lane: (K = k + 2*(lane>>4) + {0,1}, N = lane&15)  -> two b32 loads
// C-layout: reg r, lane l -> (M = r + (l>>4)*8, N = l&15)
// ---------------------------------------------------------------------------
__global__ void wmma_gemm_f32(const float* __restrict__ A, int lda, long sA,
                              const float* __restrict__ W, int ldw, long sW,
                              float* __restrict__ C, int ldc, long sC, int K) {
  const int lane = threadIdx.x & 31;
  const int wave = threadIdx.x >> 5;
  const int half = lane >> 4;
  const int r16 = lane & 15;
  const int m0 = blockIdx.x * 16;
  const int n0 = (blockIdx.y * 4 + wave) * 16;
  const int bz = blockIdx.z;

  const float* a_row =
      A + (size_t)bz * sA + (size_t)(m0 + r16) * lda + 2 * half;
  const float* w_col =
      W + (size_t)bz * sW + (size_t)(2 * half) * ldw + n0 + r16;

  v8f acc = {0.f, 0.f, 0.f, 0.f, 0.f, 0.f, 0.f, 0.f};

#pragma unroll 8
  for (int k = 0; k < K; k += 4) {
    v2f a = *(const v2f*)(a_row + k);
    const float* wp = w_col + (size_t)k * ldw;
    v2f b;
    b.x = wp[0];
    b.y = wp[ldw];
    acc = wmma4(a, b, acc);
  }

  float* c_base =
      C + (size_t)bz * sC + (size_t)(m0 + half * 8) * ldc + n0 + r16;
#pragma unroll
  for (int r = 0; r < 8; ++r) c_base[(size_t)r * ldc] = acc[r];
}

// ---------------------------------------------------------------------------
// Stage 5: C = leaky_relu(A1 @ W1 + A2 @ W2 + bias)   (all [1024,1024], f32)
// Implements cat(attn_out, output) @ lin_w without materializing the concat.
// ---------------------------------------------------------------------------
__global__ void wmma_gemm2_bias_lrelu(const float* __restrict__ A1,
                                      const float* __restrict__ W1,
                                      const float* __restrict__ A2,
                                      const float* __restrict__ W2,
                                      const float* __restrict__ bias,
                                      float* __restrict__ C, int lda, int ldw,
                                      int ldc, int K) {
  const int lane = threadIdx.x & 31;
  const int wave = threadIdx.x >> 5;
  const int half = lane >> 4;
  const int r16 = lane & 15;
  const int m0 = blockIdx.x * 16;
  const int n0 = (blockIdx.y * 4 + wave) * 16;

  v8f acc = {0.f, 0.f, 0.f, 0.f, 0.f, 0.f, 0.f, 0.f};

  {
    const float* a_row = A1 + (size_t)(m0 + r16) * lda + 2 * half;
    const float* w_col = W1 + (size_t)(2 * half) * ldw + n0 + r16;
#pragma unroll 8
    for (int k = 0; k < K; k += 4) {
      v2f a = *(const v2f*)(a_row + k);
      const float* wp = w_col + (size_t)k * ldw;
      v2f b;
      b.x = wp[0];
      b.y = wp[ldw];
      acc = wmma4(a, b, acc);
    }
  }
  {
    const float* a_row = A2 + (size_t)(m0 + r16) * lda + 2 * half;
    const float* w_col = W2 + (size_t)(2 * half) * ldw + n0 + r16;
#pragma unroll 8
    for (int k = 0; k < K; k += 4) {
      v2f a = *(const v2f*)(a_row + k);
      const float* wp = w_col + (size_t)k * ldw;
      v2f b;
      b.x = wp[0];
      b.y = wp[ldw];
      acc = wmma4(a, b, acc);
    }
  }

  const float bn = bias[n0 + r16];
  float* c_base = C + (size_t)(m0 + half * 8) * ldc + n0 + r16;
#pragma unroll
  for (int r = 0; r < 8; ++r) {
    float x = acc[r] + bn;
    c_base[(size_t)r * ldc] = x > 0.f ? x : 0.01f * x;
  }
}

// ---------------------------------------------------------------------------
// Stage 2+3 fused: score + mask + softmax -> attn [B,O,I]
// 256 threads = 8 waves; wave w owns output row (b, o_base + w).
// Per-lane register layout over D: float4 at d = j*128 + lane*4, j = 0..7.
// ---------------------------------------------------------------------------
__global__ void score_softmax(const float* __restrict__ s_out,
                              const float* __restrict__ s_ctx,
                              const float* __restrict__ w_b,
                              const float* __restrict__ score_w,
                              const float* __restrict__ score_b,
                              const int* __restrict__ mask,
                              float* __restrict__ attn) {
  const int lane = threadIdx.x & 31;
  const int wave = threadIdx.x >> 5;
  const int b = blockIdx.x >> 4;
  const int o = ((blockIdx.x & 15) << 3) + wave;

  const float4* so_p = (const float4*)(s_out + (size_t)(b * OO + o) * DD);
  const float4* wb_p = (const float4*)w_b;
  const float4* sw_p = (const float4*)score_w;

  float4 so[8], wb[8], sw[8];
#pragma unroll
  for (int j = 0; j < 8; ++j) {
    so[j] = so_p[j * 32 + lane];
    wb[j] = wb_p[j * 32 + lane];
    sw[j] = sw_p[j * 32 + lane];
  }
  int mk[4];
#pragma unroll
  for (int q = 0; q < 4; ++q) mk[q] = mask[b * II + q * 32 + lane];
  const float sb = score_b[0];

  float sc[4];
#pragma unroll
  for (int q = 0; q < 4; ++q) {
    float scq = 0.f;
    for (int ii = 0; ii < 32; ++ii) {
      const int i = q * 32 + ii;
      const float4* cx = (const float4*)(s_ctx + (size_t)(b * II + i) * DD);
      float acc = 0.f;
#pragma unroll
      for (int j = 0; j < 8; ++j) {
        float4 c4 = cx[j * 32 + lane];
        float x;
        x = so[j].x + c4.x + wb[j].x;
        acc = fmaf(x > 0.f ? x : 0.01f * x, sw[j].x, acc);
        x = so[j].y + c4.y + wb[j].y;
        acc = fmaf(x > 0.f ? x : 0.01f * x, sw[j].y, acc);
        x = so[j].z + c4.z + wb[j].z;
        acc = fmaf(x > 0.f ? x : 0.01f * x, sw[j].z, acc);
        x = so[j].w + c4.w + wb[j].w;
        acc = fmaf(x > 0.f ? x : 0.01f * x, sw[j].w, acc);
      }
#pragma unroll
      for (int off = 16; off > 0; off >>= 1) acc += __shfl_xor(acc, off, 32);
      scq = (lane == ii) ? (acc + sb) : scq;  // lane l keeps i = q*32 + l
    }
    sc[q] = scq;
  }

  // mask: score = -inf where mask == 1
#pragma unroll
  for (int q = 0; q < 4; ++q)
    if (mk[q] == 1) sc[q] = -INFINITY;

  // softmax over 128 = 4 regs x 32 lanes
  float m = fmaxf(fmaxf(sc[0], sc[1]), fmaxf(sc[2], sc[3]));
#pragma unroll
  for (int off = 16; off > 0; off >>= 1) m = fmaxf(m, __shfl_xor(m, off, 32));
  float e[4], s = 0.f;
#pragma unroll
  for (int q = 0; q < 4; ++q) {
    e[q] = __expf(sc[q] - m);
    s += e[q];
  }
#pragma unroll
  for (int off = 16; off > 0; off >>= 1) s += __shfl_xor(s, off, 32);
  const float inv = 1.0f / s;

  float* out = attn + (size_t)(b * OO + o) * II;
#pragma unroll
  for (int q = 0; q < 4; ++q) out[q * 32 + lane] = e[q] * inv;
}

// ---------------------------------------------------------------------------
extern "C" void kernel_launch(void* const* d_in, const int* in_sizes, int n_in,
                              void* d_out, int out_size, void* d_ws,
                              size_t ws_size, hipStream_t stream) {
  const float* output = (const float*)d_in[0];   // [B,O,D]
  const float* context = (const float*)d_in[1];  // [B,I,D]
  const int* mask = (const int*)d_in[2];         // [B,I]
  const float* w_out_w = (const float*)d_in[3];  // [D,D]
  const float* w_ctx_w = (const float*)d_in[4];  // [D,D]
  const float* w_b = (const float*)d_in[5];      // [D]
  const float* score_w = (const float*)d_in[6];  // [D]
  const float* score_b = (const float*)d_in[7];  // scalar
  const float* lin_w = (const float*)d_in[8];    // [2D,D]
  const float* lin_b = (const float*)d_in[9];    // [D]

  float* out_final = (float*)d_out;                  // [B,O,D]
  float* attn = out_final + (size_t)BB * OO * DD;    // [B,O,I]

  float* ws = (float*)d_ws;
  float* s_out = ws;                                 // B*O*D
  float* s_ctx = s_out + (size_t)BB * OO * DD;       // B*I*D
  float* attn_mid = s_ctx + (size_t)BB * II * DD;    // B*O*D

  dim3 blk(128);

  // Stage 1: s_out = output @ w_out_w ; s_ctx = context @ w_ctx_w
  wmma_gemm_f32<<<dim3(BB * OO / 16, DD / 64, 1), blk, 0, stream>>>(
      output, DD, 0, w_out_w, DD, 0, s_out, DD, 0, DD);
  wmma_gemm_f32<<<dim3(BB * II / 16, DD / 64, 1), blk, 0, stream>>>(
      context, DD, 0, w_ctx_w, DD, 0, s_ctx, DD, 0, DD);

  // Stage 2+3: additive-attention scores + mask + softmax -> attn
  score_softmax<<<dim3(BB * OO / 8), dim3(256), 0, stream>>>(
      s_out, s_ctx, w_b, score_w, score_b, mask, attn);

  // Stage 4: attn_mid[b] = attn[b] @ context[b]
  wmma_gemm_f32<<<dim3(OO / 16, DD / 64, BB), blk, 0, stream>>>(
      attn, II, (long)OO * II, context, DD, (long)II * DD, attn_mid, DD,
      (long)OO * DD, II);

  // Stage 5: out = lrelu(attn_mid @ lin_w[0:D,:] + output @ lin_w[D:2D,:] + b)
  wmma_gemm2_bias_lrelu<<<dim3(BB * OO / 16, DD / 64, 1), blk, 0, stream>>>(
      attn_mid, lin_w, output, lin_w + (size_t)DD * DD, lin_b, out_final, DD,
      DD, DD, DD);
}